// CompressedIndicatorEmbedding_56865366999597
// MI455X (gfx1250) — compile-verified
//
#include <hip/hip_runtime.h>
#include <hip/hip_bf16.h>
#include <stdint.h>

// CompressedIndicatorEmbedding: out[n, :] = sum_p W[:, p*512 + x[n,p]]
//   N = 1<<20 tokens, P = 8, L = 512, D = 64, W is [64, 4096] fp32 row-major.
//
// Memory-bound (out = 256 MB of write-once stores dominates; table = 1 MB).
//   pass 1: transpose W into 4 D-quarter tables with 16 B pad every 4 rows
//           (bank-conflict-free-ish gather), contiguous 272 KB per quarter.
//   pass 2: per workgroup: async-stage one quarter into LDS
//           (global_load_async_to_lds_b128 + s_wait_asynccnt), gather-sum
//           8 rows/token from LDS, non-temporal float4 stores to out
//           (keeps x and the table L2-resident across the 4 quarter passes).

#define EIF_P 8
#define EIF_L 512
#define EIF_D 64
#define EIF_N (1 << 20)
#define EIF_COLS (EIF_P * EIF_L)              // 4096 rows per quarter table

// padded quarter layout: row r of 16 floats at byte off r*64 + (r>>2)*16
#define QUARTER_BYTES (EIF_COLS * 64 + (EIF_COLS / 4) * 16)   // 278528 = 272 KB
#define QUARTER_FLOATS (QUARTER_BYTES / 4)                    // 69632

#define MAIN_THREADS 512
#define TOK_PER_BLOCK 8192
#define STAGE_ROUNDS (QUARTER_BYTES / (MAIN_THREADS * 16))    // 34 exactly

typedef float v4f __attribute__((ext_vector_type(4)));
typedef int   v4i __attribute__((ext_vector_type(4)));

// ---------------------------------------------------------------------------
// Pass 1: W[d, c] -> padded quarter-major workspace.
//   q = d/16, r = d%16; float index = q*QUARTER_FLOATS + c*16 + (c>>2)*4 + r
// ---------------------------------------------------------------------------
__global__ void eif_transpose_w(const float* __restrict__ W,
                                float* __restrict__ wq) {
    int tid = blockIdx.x * blockDim.x + threadIdx.x;   // 0 .. 64*4096-1
    if (tid >= EIF_D * EIF_COLS) return;
    int d = tid >> 12;                 // row of W (0..63), coalesced read on c
    int c = tid & (EIF_COLS - 1);
    float v = W[tid];
    int q = d >> 4;
    int r = d & 15;
    wq[q * QUARTER_FLOATS + c * 16 + (c >> 2) * 4 + r] = v;
}

// ---------------------------------------------------------------------------
// Pass 2: blockIdx.x = quarter (low 2 bits) | token-block (rest).
//   512 threads: 4 threads per token (one float4 each), 128 tokens/iteration.
// ---------------------------------------------------------------------------
__global__ void __launch_bounds__(MAIN_THREADS)
eif_gather_lds(const int* __restrict__ x,
               const float* __restrict__ wq,
               float* __restrict__ out) {
    extern __shared__ float s_w[];            // QUARTER_FLOATS floats (272 KB)

    const int q   = blockIdx.x & 3;
    const int tb  = blockIdx.x >> 2;
    const int tid = threadIdx.x;

    // ---- async stage: 272 KB contiguous, 34 rounds x 512 threads x 16 B ----
    {
        const char* src = (const char*)(wq + (size_t)q * QUARTER_FLOATS);
        unsigned lds_base = (unsigned)__builtin_amdgcn_groupstaticsize();
        #pragma unroll
        for (int r = 0; r < STAGE_ROUNDS; ++r) {
            unsigned off = (unsigned)(tid * 16 + r * (MAIN_THREADS * 16));
            asm volatile(
                "global_load_async_to_lds_b128 %0, %1, %2"
                :
                : "v"(lds_base + off),   // LDS byte offset (VDST)
                  "v"(off),              // 32-bit voffset   (VADDR, GVS mode)
                  "s"(src)               // 64-bit SGPR base (SADDR)
                : "memory");
        }
        asm volatile("s_wait_asynccnt 0x0" ::: "memory");
    }
    __syncthreads();

    // ---- gather + sum from LDS, NT store ----
    const int sub = tid & 3;                   // which float4 of the 16 dims
    const int tok_stride = MAIN_THREADS / 4;   // 128 tokens in flight
    const int n_base = tb * TOK_PER_BLOCK + (tid >> 2);
    const v4f* s4 = (const v4f*)s_w;

    for (int it = 0; it < TOK_PER_BLOCK / tok_stride; ++it) {
        const int n = n_base + it * tok_stride;

        // prefetch next iteration's index cachelines (speculative, OOB-safe)
        __builtin_prefetch(x + (size_t)(n + tok_stride) * EIF_P, 0, 1);

        const v4i* xp = (const v4i*)(x + (size_t)n * EIF_P);
        v4i a = xp[0];
        v4i b = xp[1];

        int rows[8] = { a.x,             EIF_L     + a.y,
                        2 * EIF_L + a.z, 3 * EIF_L + a.w,
                        4 * EIF_L + b.x, 5 * EIF_L + b.y,
                        6 * EIF_L + b.z, 7 * EIF_L + b.w };

        v4f acc = (v4f)(0.f);
        #pragma unroll
        for (int p = 0; p < EIF_P; ++p) {
            int row = rows[p];
            // padded row: float4 index = row*4 + (row>>2) + sub
            acc += s4[(row << 2) + (row >> 2) + sub];   // ds_load_b128
        }

        // streaming store: never re-read, keep it out of L2 (TH_NT)
        v4f* o = (v4f*)(out + (size_t)n * EIF_D + q * 16);
        __builtin_nontemporal_store(acc, o + sub);
    }
}

// ---------------------------------------------------------------------------
// Fallback (workspace too small): direct gather from W, one thread per (n,d).
// ---------------------------------------------------------------------------
__global__ void eif_gather_direct(const int* __restrict__ x,
                                  const float* __restrict__ W,
                                  float* __restrict__ out) {
    long long gid = (long long)blockIdx.x * blockDim.x + threadIdx.x;
    if (gid >= (long long)EIF_N * EIF_D) return;
    int n = (int)(gid >> 6);
    int d = (int)(gid & 63);
    const int* xr = x + (size_t)n * EIF_P;
    const float* wrow = W + (size_t)d * EIF_COLS;
    float acc = 0.f;
    #pragma unroll
    for (int p = 0; p < EIF_P; ++p) acc += wrow[p * EIF_L + xr[p]];
    out[gid] = acc;
}

// ---------------------------------------------------------------------------
extern "C" void kernel_launch(void* const* d_in, const int* in_sizes, int n_in,
                              void* d_out, int out_size, void* d_ws, size_t ws_size,
                              hipStream_t stream) {
    const int*   x = (const int*)d_in[0];     // [N, 8] int32
    const float* W = (const float*)d_in[1];   // [64, 4096] fp32
    float*     out = (float*)d_out;           // [N, 64] fp32

    if (ws_size >= (size_t)4 * QUARTER_BYTES) {
        float* wq = (float*)d_ws;
        // pass 1: ~1 MB transpose into padded quarter-major layout
        eif_transpose_w<<<(EIF_D * EIF_COLS) / 256, 256, 0, stream>>>(W, wq);
        // pass 2: 4 quarters x 128 token-blocks, 272 KB dynamic LDS each
        int nblocks = 4 * (EIF_N / TOK_PER_BLOCK);
        eif_gather_lds<<<nblocks, MAIN_THREADS, QUARTER_BYTES, stream>>>(x, wq, out);
    } else {
        long long total = (long long)EIF_N * EIF_D;
        int nblocks = (int)((total + 255) / 256);
        eif_gather_direct<<<nblocks, 256, 0, stream>>>(x, W, out);
    }
}